// FeatureSeperationLoss_53137335386661
// MI455X (gfx1250) — compile-verified
//
#include <hip/hip_runtime.h>
#include <math.h>

// ---- problem constants (from reference) ------------------------------------
#define NUM_CLASS 19
#define C_DIM 256
#define H_DIM 128
#define W_DIM 128
#define B_DIM 8
#define HW (H_DIM * W_DIM)              // 16384
#define NBH (B_DIM * H_DIM)             // 1024 (b,h) rows
#define ROWS_PER_BLOCK 4
#define MAIN_BLOCKS (NBH / ROWS_PER_BLOCK)  // 256 blocks
#define LDS_STRIDE 257                  // 256 + 1 pad -> distinct LDS banks per class row
#define SUM_ELEMS (NUM_CLASS * C_DIM)   // 4864
#define WS_ELEMS (SUM_ELEMS + NUM_CLASS)

typedef __attribute__((ext_vector_type(2))) float v2f;
typedef __attribute__((ext_vector_type(8))) float v8f;

// ---- kernel 0: zero the workspace accumulators -----------------------------
__global__ void zero_ws_kernel(float* __restrict__ ws) {
  int i = blockIdx.x * blockDim.x + threadIdx.x;
  if (i < WS_ELEMS) ws[i] = 0.0f;
}

// ---- kernel 1: bandwidth-bound segment sum ---------------------------------
// 128 MiB of features streamed once with b128 loads; per-block partials in LDS
// via ds_add_f32; one global_atomic_add_f32 flush per block.
__global__ __launch_bounds__(256) void segsum_kernel(const float* __restrict__ feat,
                                                     const int* __restrict__ labels,
                                                     float* __restrict__ gsum,
                                                     float* __restrict__ gcnt) {
  __shared__ float lsum[NUM_CLASS * LDS_STRIDE];
  __shared__ float lcnt[NUM_CLASS];

  const int tid = threadIdx.x;
  for (int i = tid; i < NUM_CLASS * LDS_STRIDE; i += 256) lsum[i] = 0.0f;
  if (tid < NUM_CLASS) lcnt[tid] = 0.0f;
  __syncthreads();

  const int w4 = (tid & 31) << 2;  // 4 pixels per lane along W: coalesced b128
  const int cg = tid >> 5;         // 8 channel groups (one per wave)

  for (int r = 0; r < ROWS_PER_BLOCK; ++r) {
    const int bh = blockIdx.x * ROWS_PER_BLOCK + r;
    const int b = bh >> 7;
    const int h = bh & (H_DIM - 1);

    const int4 lab = *reinterpret_cast<const int4*>(&labels[bh * W_DIM + w4]);
    const float* fbase = feat + (size_t)b * (size_t)(C_DIM * HW) + (size_t)(h * W_DIM + w4);

    for (int c = cg; c < C_DIM; c += 8) {
      const float4 v = *reinterpret_cast<const float4*>(&fbase[(size_t)c * HW]);
      __hip_atomic_fetch_add(&lsum[lab.x * LDS_STRIDE + c], v.x, __ATOMIC_RELAXED, __HIP_MEMORY_SCOPE_WORKGROUP);
      __hip_atomic_fetch_add(&lsum[lab.y * LDS_STRIDE + c], v.y, __ATOMIC_RELAXED, __HIP_MEMORY_SCOPE_WORKGROUP);
      __hip_atomic_fetch_add(&lsum[lab.z * LDS_STRIDE + c], v.z, __ATOMIC_RELAXED, __HIP_MEMORY_SCOPE_WORKGROUP);
      __hip_atomic_fetch_add(&lsum[lab.w * LDS_STRIDE + c], v.w, __ATOMIC_RELAXED, __HIP_MEMORY_SCOPE_WORKGROUP);
    }
    if (cg == 0) {  // count each pixel exactly once
      __hip_atomic_fetch_add(&lcnt[lab.x], 1.0f, __ATOMIC_RELAXED, __HIP_MEMORY_SCOPE_WORKGROUP);
      __hip_atomic_fetch_add(&lcnt[lab.y], 1.0f, __ATOMIC_RELAXED, __HIP_MEMORY_SCOPE_WORKGROUP);
      __hip_atomic_fetch_add(&lcnt[lab.z], 1.0f, __ATOMIC_RELAXED, __HIP_MEMORY_SCOPE_WORKGROUP);
      __hip_atomic_fetch_add(&lcnt[lab.w], 1.0f, __ATOMIC_RELAXED, __HIP_MEMORY_SCOPE_WORKGROUP);
    }
  }
  __syncthreads();

  for (int i = tid; i < SUM_ELEMS; i += 256) {
    const int k = i >> 8;
    const int c = i & 255;
    __hip_atomic_fetch_add(&gsum[i], lsum[k * LDS_STRIDE + c], __ATOMIC_RELAXED, __HIP_MEMORY_SCOPE_AGENT);
  }
  if (tid < NUM_CLASS)
    __hip_atomic_fetch_add(&gcnt[tid], lcnt[tid], __ATOMIC_RELAXED, __HIP_MEMORY_SCOPE_AGENT);
}

// ---- kernel 2: means -> normalize -> WMMA gram -> loss (one wave32) --------
__global__ __launch_bounds__(32) void finalize_kernel(const float* __restrict__ gsum,
                                                      const float* __restrict__ gcnt,
                                                      float* __restrict__ out) {
  __shared__ float mn[32 * LDS_STRIDE];  // 19 real rows padded to 32 with zeros
  const int tid = threadIdx.x;

  for (int i = tid; i < 32 * C_DIM; i += 32) {
    const int k = i >> 8;
    const int c = i & 255;
    float v = 0.0f;
    if (k < NUM_CLASS) v = gsum[k * C_DIM + c] / fmaxf(gcnt[k], 1.0f);
    mn[k * LDS_STRIDE + c] = v;
  }
  __syncthreads();

  if (tid < NUM_CLASS) {
    float ss = 0.0f;
    for (int c = 0; c < C_DIM; ++c) {
      const float m = mn[tid * LDS_STRIDE + c];
      ss += m * m;
    }
    const float inv = 1.0f / fmaxf(sqrtf(ss), 1e-12f);
    for (int c = 0; c < C_DIM; ++c) mn[tid * LDS_STRIDE + c] *= inv;
  }
  __syncthreads();

  // G(32x32) = mn(32x256) @ mn^T(256x32) via V_WMMA_F32_16X16X4_F32.
  // f32 A fragment (16x4): lanes 0-15 hold K=k0+{0,1}, lanes 16-31 hold K=k0+{2,3};
  // by symmetry of the gram the same fragment serves as the B operand.
  const int r0 = tid & 15;
  const int koff = (tid >> 4) << 1;  // 0 or 2
  v8f acc[2][2] = {{{0.f}, {0.f}}, {{0.f}, {0.f}}};

  for (int k0 = 0; k0 < C_DIM; k0 += 4) {
    v2f a0, a1;
    a0.x = mn[r0 * LDS_STRIDE + k0 + koff];
    a0.y = mn[r0 * LDS_STRIDE + k0 + koff + 1];
    a1.x = mn[(r0 + 16) * LDS_STRIDE + k0 + koff];
    a1.y = mn[(r0 + 16) * LDS_STRIDE + k0 + koff + 1];
    acc[0][0] = __builtin_amdgcn_wmma_f32_16x16x4_f32(false, a0, false, a0, (short)0, acc[0][0], false, false);
    acc[0][1] = __builtin_amdgcn_wmma_f32_16x16x4_f32(false, a0, false, a1, (short)0, acc[0][1], false, false);
    acc[1][0] = __builtin_amdgcn_wmma_f32_16x16x4_f32(false, a1, false, a0, (short)0, acc[1][0], false, false);
    acc[1][1] = __builtin_amdgcn_wmma_f32_16x16x4_f32(false, a1, false, a1, (short)0, acc[1][1], false, false);
  }

  // D = 1 - G off-diag (relu(0.5 - D) = relu(g - 0.5)); diagonal forced to D=2 -> 0.
  // C/D layout: VGPR r -> M = base + r (+8 for lanes 16-31), lane -> N.
  float partial = 0.0f;
  const int ncol = tid & 15;
  const int mhalf = (tid >> 4) << 3;
#pragma unroll
  for (int i = 0; i < 2; ++i)
#pragma unroll
    for (int j = 0; j < 2; ++j)
#pragma unroll
      for (int rr = 0; rr < 8; ++rr) {
        const int M = i * 16 + mhalf + rr;
        const int N = j * 16 + ncol;
        if (M < NUM_CLASS && N < NUM_CLASS && M != N)
          partial += fmaxf(acc[i][j][rr] - 0.5f, 0.0f);
      }

  for (int off = 16; off > 0; off >>= 1)
    partial += __shfl_down(partial, off, 32);
  if (tid == 0) out[0] = partial * (1.0f / (float)(NUM_CLASS * NUM_CLASS));
}

// ---- host-side launcher ----------------------------------------------------
extern "C" void kernel_launch(void* const* d_in, const int* in_sizes, int n_in,
                              void* d_out, int out_size, void* d_ws, size_t ws_size,
                              hipStream_t stream) {
  const float* feat = (const float*)d_in[0];   // (8,256,128,128) f32
  const int* labels = (const int*)d_in[1];     // (8,128,128) i32
  // d_in[2] = prototypes: unused by the reference computation.
  float* ws = (float*)d_ws;
  float* gsum = ws;                 // 19*256 f32
  float* gcnt = ws + SUM_ELEMS;     // 19 f32
  float* out = (float*)d_out;

  zero_ws_kernel<<<(WS_ELEMS + 255) / 256, 256, 0, stream>>>(ws);
  segsum_kernel<<<MAIN_BLOCKS, 256, 0, stream>>>(feat, labels, gsum, gcnt);
  finalize_kernel<<<1, 32, 0, stream>>>(gsum, gcnt, out);
}